// GCN_11295763988681
// MI455X (gfx1250) — compile-verified
//
#include <hip/hip_runtime.h>
#include <hip/hip_bf16.h>

typedef __attribute__((ext_vector_type(2))) float v2f;
typedef __attribute__((ext_vector_type(8))) float v8f;

#define NUMK 128
#define NUME 1024
#define HID  32
#define HXW  (NUMK + NUME + 2)   // 1154
#define AS   132                 // A row stride: bank step 4, conflict-free A-operand loads
#define XS   36                  // x1/x2 row stride: {36l mod 64} distinct for 16 lanes
#define GTS  132                 // transposed-G column stride (k-contiguous rows)
#define WTS  36                  // transposed-W2 column stride

// D = A(16x4) x B(4x16) + C, f32 WMMA, all operands sourced from LDS.
// A layout: lane l<16 holds A[m0+l][k0+0 / k0+1] in v0/v1; lanes>=16 hold K=k0+2 / k0+3.
// B layout: v0 = row k0 (lanes 0-15) / row k0+2 (lanes 16-31); v1 = rows k0+1 / k0+3.
//           B is stored transposed (BT[n][k], k contiguous) so each lane's (k,k+1) pair
//           is one 8-byte LDS load, same as the A side.
// C/D layout: vgpr j: lane l<16 -> [m0+j][n0+l]; lanes>=16 -> [m0+8+j][n0+l].
__device__ __forceinline__ v8f wmma_step(const float* __restrict__ Am, int aStride,
                                         const float* __restrict__ BTm, int btStride,
                                         int m0, int n0, int k0, int half, int l, v8f c) {
    v2f a, b;
    a.x = Am[(m0 + l) * aStride + k0 + 2 * half];
    a.y = Am[(m0 + l) * aStride + k0 + 2 * half + 1];
    b.x = BTm[(n0 + l) * btStride + k0 + 2 * half];
    b.y = BTm[(n0 + l) * btStride + k0 + 2 * half + 1];
    return __builtin_amdgcn_wmma_f32_16x16x4_f32(false, a, false, b, (short)0, c,
                                                 false, false);
}

__global__ __launch_bounds__(256) void gcn3_kernel(
    const float* __restrict__ Hx, const int* __restrict__ ei,
    const float* __restrict__ W1, const float* __restrict__ b1,
    const float* __restrict__ W2, const float* __restrict__ b2,
    const float* __restrict__ W3, const float* __restrict__ b3,
    float* __restrict__ out) {
    __shared__ float As[NUMK * AS];     // dense normalized adjacency (row = dst)
    __shared__ float Xs[NUMK * XS];     // x1 (A-operand), then x2 (row-major)
    __shared__ float GT[HID * GTS];     // (x1 @ W2) transposed: GT[n][k]
    __shared__ float W2T[HID * WTS];    // W2 transposed: W2T[n][k]
    __shared__ float ps[NUMK], a1s[NUMK], vs[NUMK], deg[NUMK], dinv[NUMK];
    __shared__ float W1s[HID], b1s[HID], b2s[HID], W3s[HID];

    const int g    = blockIdx.x;
    const int t    = threadIdx.x;        // 0..255 (8 wave32)
    const int lane = t & 31;
    const int wave = t >> 5;
    const int half = lane >> 4;
    const int l    = lane & 15;
    const float* hx = Hx + (size_t)g * HXW;

    // ---- phase 0: zero A, stage params + node signals ----
    for (int i = t; i < NUMK * AS; i += 256) As[i] = 0.0f;
    if (t < NUMK) ps[t] = hx[t];
    if (t < HID) { W1s[t] = W1[t]; b1s[t] = b1[t]; b2s[t] = b2[t]; W3s[t] = W3[t]; }
    for (int i = t; i < HID * HID; i += 256)
        W2T[(i & 31) * WTS + (i >> 5)] = W2[i];   // W2 row-major [k][n] -> W2T[n][k]
    __syncthreads();

    // ---- phase 1: self-loops ----
    if (t < NUMK) { As[t * AS + t] = 1.0f; deg[t] = 1.0f; }
    __syncthreads();

    // ---- phase 2: scatter edges (duplicates accumulate, like segment_sum) ----
    for (int e = t; e < NUME; e += 256) {
        int s = ei[e];
        int d = ei[NUME + e];
        float w = hx[NUMK + e];
        atomicAdd(&As[d * AS + s], w);   // ds_add_f32
        atomicAdd(&deg[d], w);
    }
    __syncthreads();

    // ---- phase 3: symmetric normalization (deg >= 1 always: self-loop + w>=0) ----
    if (t < NUMK) dinv[t] = rsqrtf(deg[t]);
    __syncthreads();
    for (int i = t; i < NUMK * NUMK; i += 256) {
        int d = i >> 7, s = i & 127;
        As[d * AS + s] *= dinv[d] * dinv[s];
    }
    __syncthreads();

    // ---- layer 1 (rank-1): a1 = A @ p ; x1 = relu(a1 (x) W1 + b1) ----
    if (t < NUMK) {
        float acc = 0.0f;
        for (int s = 0; s < NUMK; ++s) acc += As[t * AS + s] * ps[s];
        a1s[t] = acc;
    }
    __syncthreads();
    for (int i = t; i < NUMK * HID; i += 256) {
        int n = i >> 5, f = i & 31;
        Xs[n * XS + f] = fmaxf(fmaf(a1s[n], W1s[f], b1s[f]), 0.0f);
    }
    __syncthreads();

    // ---- layer 2a: g = x1 @ W2  (WMMA, K=32), stored transposed into GT ----
    {
        const int m0 = wave * 16;
        for (int nt = 0; nt < 2; ++nt) {
            const int n0 = nt * 16;
            v8f c = {};
#pragma unroll
            for (int k0 = 0; k0 < HID; k0 += 4)
                c = wmma_step(Xs, XS, W2T, WTS, m0, n0, k0, half, l, c);
#pragma unroll
            for (int j = 0; j < 8; ++j)
                GT[(n0 + l) * GTS + (m0 + j + 8 * half)] = c[j];
        }
    }
    __syncthreads();

    // ---- layer 2b: x2 = relu(A @ g + b2)  (WMMA, K=128) ----
    {
        const int m0 = wave * 16;
        for (int nt = 0; nt < 2; ++nt) {
            const int n0 = nt * 16;
            v8f c = {};
#pragma unroll
            for (int k0 = 0; k0 < NUMK; k0 += 4)
                c = wmma_step(As, AS, GT, GTS, m0, n0, k0, half, l, c);
#pragma unroll
            for (int j = 0; j < 8; ++j) {
                int row = m0 + j + 8 * half, col = n0 + l;
                Xs[row * XS + col] = fmaxf(c[j] + b2s[col], 0.0f);
            }
        }
    }
    __syncthreads();

    // ---- layer 3 (rank-1): v = x2 @ W3 ; out = A @ v + b3 ----
    if (t < NUMK) {
        float acc = 0.0f;
        for (int f = 0; f < HID; ++f) acc += Xs[t * XS + f] * W3s[f];
        vs[t] = acc;
    }
    __syncthreads();
    if (t < NUMK) {
        float acc = 0.0f;
        for (int s = 0; s < NUMK; ++s) acc += As[t * AS + s] * vs[s];
        out[(size_t)g * NUMK + t] = acc + b3[0];
    }
}

extern "C" void kernel_launch(void* const* d_in, const int* in_sizes, int n_in,
                              void* d_out, int out_size, void* d_ws, size_t ws_size,
                              hipStream_t stream) {
    (void)n_in; (void)d_ws; (void)ws_size; (void)out_size;
    const float* Hx = (const float*)d_in[0];
    const int*   ei = (const int*)d_in[1];
    const float* W1 = (const float*)d_in[2];
    const float* b1 = (const float*)d_in[3];
    const float* W2 = (const float*)d_in[4];
    const float* b2 = (const float*)d_in[5];
    const float* W3 = (const float*)d_in[6];
    const float* b3 = (const float*)d_in[7];
    float* out = (float*)d_out;
    const int nb = in_sizes[0] / HXW;   // 2048 graphs, one workgroup each
    gcn3_kernel<<<dim3(nb), dim3(256), 0, stream>>>(Hx, ei, W1, b1, W2, b2, W3, b3, out);
}